// GAT_3951369912441
// MI455X (gfx1250) — compile-verified
//
#include <hip/hip_runtime.h>

#define NEG_SLOPE 0.2f
#define EPSF 1e-16f

typedef __attribute__((ext_vector_type(2))) float v2f;
typedef __attribute__((ext_vector_type(8))) float v8f;

__device__ __forceinline__ float lrelu(float a) { return a > 0.f ? a : NEG_SLOPE * a; }

// Race-safe float atomic max via sign-split integer atomics.
__device__ __forceinline__ void atomicMaxF(float* addr, float v) {
    if (v >= 0.f) atomicMax((int*)addr, __float_as_int(v));
    else          atomicMin((unsigned int*)addr, (unsigned int)__float_as_int(v));
}

// ---------------------------------------------------------------------------
// K0: split int64 edge_index [2,E] into int32 src[],dst[]
// ---------------------------------------------------------------------------
__global__ void cvt_edges(const long long* __restrict__ ei, int* __restrict__ src,
                          int* __restrict__ dst, int E) {
    int t = blockIdx.x * blockDim.x + threadIdx.x;
    if (t < E) {
        src[t] = (int)ei[t];
        dst[t] = (int)ei[(long long)E + t];
    }
}

// ---------------------------------------------------------------------------
// Layer-1 transform: h1[N,32] = x[N,3] @ W1[3,32]; one 16-node tile per wave,
// two V_WMMA_F32_16X16X4_F32 (cols 0-15 / 16-31), K=3 zero-padded to 4.
// All operand selection is branchless (cndmask) so EXEC stays full; stores use
// a wave-uniform full-tile fast path (no per-row exec toggling).
// ---------------------------------------------------------------------------
__global__ void l1_transform(const float* __restrict__ x, const float* __restrict__ W1,
                             float* __restrict__ h1, int N) {
    int wave = threadIdx.x >> 5;
    int lane = threadIdx.x & 31;
    int tile = (blockIdx.x * (blockDim.x >> 5) + wave) * 16;
    if (tile >= N) return;                 // wave-uniform
    int r    = lane & 15;
    int sel  = lane >> 4;                  // 0: K0/K1, 1: K2/K3
    int node = tile + r;
    int nc   = node < N ? node : N - 1;    // clamp (tail rows never stored)
    float x0 = x[nc * 3 + 0], x1 = x[nc * 3 + 1], x2v = x[nc * 3 + 2];
    v2f a;
    a.x = sel ? x2v : x0;
    a.y = sel ? 0.f : x1;                  // K=3 pad
    bool full = (tile + 16 <= N);          // wave-uniform
#pragma unroll
    for (int half = 0; half < 2; ++half) {
        int j = (lane & 15) + half * 16;   // output column
        float w0 = W1[0 * 32 + j];         // unconditional loads, valid addrs
        float w1 = W1[1 * 32 + j];
        float w2 = W1[2 * 32 + j];
        v2f b;
        b.x = sel ? w2  : w0;
        b.y = sel ? 0.f : w1;
        v8f c = {};
        c = __builtin_amdgcn_wmma_f32_16x16x4_f32(false, a, false, b, (short)0, c,
                                                  false, false);
        int rb = tile + (sel ? 8 : 0);
        if (full) {
#pragma unroll
            for (int i = 0; i < 8; ++i) h1[(rb + i) * 32 + j] = c[i];
        } else {
#pragma unroll
            for (int i = 0; i < 8; ++i)
                if (rb + i < N) h1[(rb + i) * 32 + j] = c[i];
        }
    }
}

// ---------------------------------------------------------------------------
// Layer-2 transform: h2[N,14] = x2[N,32] @ W2[32,14]; 8 chained accumulating
// V_WMMA_F32_16X16X4_F32; B columns >=14 are zeroed branchlessly.
// ---------------------------------------------------------------------------
__global__ void l2_transform(const float* __restrict__ x2, const float* __restrict__ W2,
                             float* __restrict__ h2, int N) {
    int wave = threadIdx.x >> 5;
    int lane = threadIdx.x & 31;
    int tile = (blockIdx.x * (blockDim.x >> 5) + wave) * 16;
    if (tile >= N) return;
    int r    = lane & 15;
    int sel  = lane >> 4;
    int node = tile + r;
    int nc   = node < N ? node : N - 1;
    int j    = lane & 15;                  // output column (valid < 14)
    int jc   = j < 14 ? j : 13;            // clamped column for safe loads
    bool jv  = (j < 14);
    v8f c = {};
#pragma unroll
    for (int kk = 0; kk < 8; ++kk) {
        int k0 = kk * 4 + (sel ? 2 : 0);
        v2f a;
        a.x = x2[nc * 32 + k0];
        a.y = x2[nc * 32 + k0 + 1];
        float wb0 = W2[k0 * 14 + jc];      // unconditional, clamped address
        float wb1 = W2[(k0 + 1) * 14 + jc];
        v2f b;
        b.x = jv ? wb0 : 0.f;
        b.y = jv ? wb1 : 0.f;
        c = __builtin_amdgcn_wmma_f32_16x16x4_f32(false, a, false, b, (short)0, c,
                                                  false, false);
    }
    bool full = (tile + 16 <= N);
    if (jv) {
        int rb = tile + (sel ? 8 : 0);
        if (full) {
#pragma unroll
            for (int i = 0; i < 8; ++i) h2[(rb + i) * 14 + j] = c[i];
        } else {
#pragma unroll
            for (int i = 0; i < 8; ++i)
                if (rb + i < N) h2[(rb + i) * 14 + j] = c[i];
        }
    }
}

// ---------------------------------------------------------------------------
// Per-node attention scalars + amax seeded with the self-loop edge (i->i).
// as/ad/amax stored as float2 per node for 8-byte vector access everywhere.
// ---------------------------------------------------------------------------
template <int C>
__global__ void attn_node(const float* __restrict__ h, const float* __restrict__ att_s,
                          const float* __restrict__ att_d, float2* __restrict__ as,
                          float2* __restrict__ ad, float2* __restrict__ amax, int N) {
    int n = blockIdx.x * blockDim.x + threadIdx.x;
    if (n >= N) return;
    float s[2], d[2];
#pragma unroll
    for (int hd = 0; hd < 2; ++hd) {
        s[hd] = 0.f; d[hd] = 0.f;
#pragma unroll
        for (int c = 0; c < C; ++c) {
            float v = h[n * (2 * C) + hd * C + c];
            s[hd] += v * att_s[hd * C + c];
            d[hd] += v * att_d[hd * C + c];
        }
    }
    as[n]   = make_float2(s[0], s[1]);
    ad[n]   = make_float2(d[0], d[1]);
    amax[n] = make_float2(lrelu(s[0] + d[0]), lrelu(s[1] + d[1]));  // self-loop
}

// Edge pass 1: segment max of leaky-relu logits into amax[dst] (b64 gathers)
__global__ void edge_max(const int* __restrict__ src, const int* __restrict__ dst,
                         const float2* __restrict__ as, const float2* __restrict__ ad,
                         float2* amax, int E) {
    int e = blockIdx.x * blockDim.x + threadIdx.x;
    if (e >= E) return;
    int s = src[e], d = dst[e];
    float2 a_s = as[s], a_d = ad[d];
    atomicMaxF(&((float*)&amax[d])[0], lrelu(a_s.x + a_d.x));
    atomicMaxF(&((float*)&amax[d])[1], lrelu(a_s.y + a_d.y));
}

// Node init: seed ssum/acc with the self-loop contribution (needs final amax)
template <int C>
__global__ void node_init(const float* __restrict__ h, const float2* __restrict__ as,
                          const float2* __restrict__ ad, const float2* __restrict__ amax,
                          float2* __restrict__ ssum, float* __restrict__ acc, int N) {
    int n = blockIdx.x * blockDim.x + threadIdx.x;
    if (n >= N) return;
    float2 a_s = as[n], a_d = ad[n], mx = amax[n];
    float e0 = __expf(lrelu(a_s.x + a_d.x) - mx.x);
    float e1 = __expf(lrelu(a_s.y + a_d.y) - mx.y);
    ssum[n] = make_float2(e0, e1);
#pragma unroll
    for (int c = 0; c < C; ++c) {
        acc[n * (2 * C) + c]     = h[n * (2 * C) + c]     * e0;
        acc[n * (2 * C) + C + c] = h[n * (2 * C) + C + c] * e1;
    }
}

// ---------------------------------------------------------------------------
// Edge pass 2 (fused): e = exp(logit - amax[dst]); ssum[dst]+=e;
// acc[dst] += h[src]*e.  Layer 1 (C=16): h row is 128B -> 8x b128 loads.
// Layer 2 (C=7): h row is 56B, 8B-aligned -> 7x b64 loads.
// ---------------------------------------------------------------------------
__global__ void edge_aggr16(const int* __restrict__ src, const int* __restrict__ dst,
                            const float2* __restrict__ as, const float2* __restrict__ ad,
                            const float2* __restrict__ amax, const float4* __restrict__ h4,
                            float2* ssum, float* acc, int E) {
    int e = blockIdx.x * blockDim.x + threadIdx.x;
    if (e >= E) return;
    int s = src[e], d = dst[e];
    float2 a_s = as[s], a_d = ad[d], mx = amax[d];
    float ex0 = __expf(lrelu(a_s.x + a_d.x) - mx.x);
    float ex1 = __expf(lrelu(a_s.y + a_d.y) - mx.y);
    atomicAdd(&((float*)&ssum[d])[0], ex0);
    atomicAdd(&((float*)&ssum[d])[1], ex1);
    float* accd = acc + (size_t)d * 32;
#pragma unroll
    for (int q = 0; q < 8; ++q) {                  // h row: 8 x float4
        float4 hv = h4[(size_t)s * 8 + q];
        float  ex = (q < 4) ? ex0 : ex1;           // first 4 quads = head0
        atomicAdd(accd + q * 4 + 0, hv.x * ex);
        atomicAdd(accd + q * 4 + 1, hv.y * ex);
        atomicAdd(accd + q * 4 + 2, hv.z * ex);
        atomicAdd(accd + q * 4 + 3, hv.w * ex);
    }
}

__global__ void edge_aggr7(const int* __restrict__ src, const int* __restrict__ dst,
                           const float2* __restrict__ as, const float2* __restrict__ ad,
                           const float2* __restrict__ amax, const float2* __restrict__ h2v,
                           float2* ssum, float* acc, int E) {
    int e = blockIdx.x * blockDim.x + threadIdx.x;
    if (e >= E) return;
    int s = src[e], d = dst[e];
    float2 a_s = as[s], a_d = ad[d], mx = amax[d];
    float ex0 = __expf(lrelu(a_s.x + a_d.x) - mx.x);
    float ex1 = __expf(lrelu(a_s.y + a_d.y) - mx.y);
    atomicAdd(&((float*)&ssum[d])[0], ex0);
    atomicAdd(&((float*)&ssum[d])[1], ex1);
    float* accd = acc + (size_t)d * 14;
#pragma unroll
    for (int q = 0; q < 7; ++q) {                  // h row: 7 x float2
        float2 hv = h2v[(size_t)s * 7 + q];
        int   c0 = q * 2, c1 = q * 2 + 1;          // flat cols 0..13
        float exa = (c0 < 7) ? ex0 : ex1;
        float exb = (c1 < 7) ? ex0 : ex1;
        atomicAdd(accd + c0, hv.x * exa);
        atomicAdd(accd + c1, hv.y * exb);
    }
}

// Layer-1 finalize (in place): acc/(ssum+eps) + b1, ReLU -> x2 (concat layout)
__global__ void l1_finalize(float* acc, const float2* __restrict__ ssum,
                            const float* __restrict__ b1, int N) {
    int t = blockIdx.x * blockDim.x + threadIdx.x;
    int n = t >> 5, j = t & 31;
    if (n >= N) return;
    float2 sv = ssum[n];
    float  sc = (j < 16 ? sv.x : sv.y) + EPSF;
    float  v  = acc[n * 32 + j] / sc + b1[j];
    acc[n * 32 + j] = v > 0.f ? v : 0.f;
}

// Layer-2 finalize: normalize, mean over heads, +b2, log_softmax -> out[N,7]
__global__ void l2_finalize(const float* __restrict__ acc, const float2* __restrict__ ssum,
                            const float* __restrict__ b2, float* __restrict__ out, int N) {
    int n = blockIdx.x * blockDim.x + threadIdx.x;
    if (n >= N) return;
    float2 sv = ssum[n];
    float i0 = 1.f / (sv.x + EPSF);
    float i1 = 1.f / (sv.y + EPSF);
    float v[7], mx = -3.0e38f;
#pragma unroll
    for (int c = 0; c < 7; ++c) {
        float a = acc[n * 14 + c] * i0;
        float b = acc[n * 14 + 7 + c] * i1;
        v[c] = 0.5f * (a + b) + b2[c];
        mx = fmaxf(mx, v[c]);
    }
    float se = 0.f;
#pragma unroll
    for (int c = 0; c < 7; ++c) se += __expf(v[c] - mx);
    float lse = __logf(se) + mx;
#pragma unroll
    for (int c = 0; c < 7; ++c) out[n * 7 + c] = v[c] - lse;
}

// ---------------------------------------------------------------------------
extern "C" void kernel_launch(void* const* d_in, const int* in_sizes, int n_in,
                              void* d_out, int out_size, void* d_ws, size_t ws_size,
                              hipStream_t stream) {
    const float*     x        = (const float*)d_in[0];
    const long long* ei       = (const long long*)d_in[1];
    const float*     W1       = (const float*)d_in[2];
    const float*     att_src1 = (const float*)d_in[3];
    const float*     att_dst1 = (const float*)d_in[4];
    const float*     b1       = (const float*)d_in[5];
    const float*     W2       = (const float*)d_in[6];
    const float*     att_src2 = (const float*)d_in[7];
    const float*     att_dst2 = (const float*)d_in[8];
    const float*     b2       = (const float*)d_in[9];
    float*           out      = (float*)d_out;

    const int N = in_sizes[0] / 3;
    const int E = in_sizes[1] / 2;

    // ---- workspace carve-up (16B-aligned regions; all written before read) --
    char* ws = (char*)d_ws;
    size_t off = 0;
    int*    src32 = (int*)(ws + off);    off += (size_t)E * 4;
    int*    dst32 = (int*)(ws + off);    off += (size_t)E * 4;
    float*  h1    = (float*)(ws + off);  off += (size_t)N * 32 * 4;  // reused as h2
    float2* as    = (float2*)(ws + off); off += (size_t)N * 8;
    float2* ad    = (float2*)(ws + off); off += (size_t)N * 8;
    float2* amax  = (float2*)(ws + off); off += (size_t)N * 8;
    float2* ssum  = (float2*)(ws + off); off += (size_t)N * 8;
    float*  acc1  = (float*)(ws + off);  off += (size_t)N * 32 * 4;  // -> x2 in place
    float*  acc2  = (float*)(ws + off);  off += (size_t)N * 14 * 4;
    (void)ws_size;
    float* h2 = h1;       // layer-1 h dead once layer-2 transform has run
    float* x2 = acc1;     // l1_finalize normalizes acc1 in place

    const int B = 256;
    const int gE   = (E + B - 1) / B;
    const int gN   = (N + B - 1) / B;
    const int gT   = (N + 127) / 128;       // 8 waves/block * 16 nodes/wave
    const int gN32 = ((N * 32) + B - 1) / B;

    cvt_edges<<<gE, B, 0, stream>>>(ei, src32, dst32, E);

    // ---------------- layer 1: 3 -> 2x16, concat, relu ----------------
    l1_transform<<<gT, B, 0, stream>>>(x, W1, h1, N);
    attn_node<16><<<gN, B, 0, stream>>>(h1, att_src1, att_dst1, as, ad, amax, N);
    edge_max<<<gE, B, 0, stream>>>(src32, dst32, as, ad, amax, E);
    node_init<16><<<gN, B, 0, stream>>>(h1, as, ad, amax, ssum, acc1, N);
    edge_aggr16<<<gE, B, 0, stream>>>(src32, dst32, as, ad, amax,
                                      (const float4*)h1, ssum, acc1, E);
    l1_finalize<<<gN32, B, 0, stream>>>(acc1, ssum, b1, N);

    // ---------------- layer 2: 32 -> 2x7, mean, log_softmax ----------------
    l2_transform<<<gT, B, 0, stream>>>(x2, W2, h2, N);
    attn_node<7><<<gN, B, 0, stream>>>(h2, att_src2, att_dst2, as, ad, amax, N);
    edge_max<<<gE, B, 0, stream>>>(src32, dst32, as, ad, amax, E);
    node_init<7><<<gN, B, 0, stream>>>(h2, as, ad, amax, ssum, acc2, N);
    edge_aggr7<<<gE, B, 0, stream>>>(src32, dst32, as, ad, amax,
                                     (const float2*)h2, ssum, acc2, E);
    l2_finalize<<<gN, B, 0, stream>>>(acc2, ssum, b2, out, N);
}